// VectorQuantizer_28509992911145
// MI455X (gfx1250) — compile-verified
//
#include <hip/hip_runtime.h>
#include <math.h>

typedef __attribute__((ext_vector_type(2))) float        v2f;
typedef __attribute__((ext_vector_type(8))) float        v8f;
typedef __attribute__((ext_vector_type(4))) unsigned int u32x4;
typedef __attribute__((ext_vector_type(8))) int          i32x8;
typedef __attribute__((ext_vector_type(4))) int          i32x4;

#define DIM   64
#define KCB   512
#define NROWS 65536
#define ROWS_PER_WAVE   16
#define WAVES_PER_BLOCK 4
#define NTILES      (KCB / 16)   // 32 code tiles
#define CHUNK_TILES 4            // tiles per TDM chunk (16 KB)
#define NCHUNKS     (NTILES / CHUNK_TILES)
#define CHUNK_F4    (CHUNK_TILES * 8 * 32)   // 1024 float4 = 16 KB

#if defined(__has_builtin)
#if __has_builtin(__builtin_amdgcn_tensor_load_to_lds) && \
    __has_builtin(__builtin_amdgcn_s_wait_tensorcnt)
#define HAVE_TDM 1
#endif
#endif

// ---------------------------------------------------------------------------
// Kernel 0: pack codebook into WMMA-B register layout (float4 = 2 k-slices),
// compute w2[k] = sum_d w[d,k]^2, zero histogram + SSE.
// wpack index: ((ct*8 + sp)*32 + lane) ; lane = half*16 + col%16 ; koff=2*half
//   f4 = { w[(8sp+koff)K+kcol], w[(8sp+koff+1)K+kcol],
//          w[(8sp+4+koff)K+kcol], w[(8sp+4+koff+1)K+kcol] }
// Launch: 16 blocks x 512 threads (8192 float4 total = 128 KB).
// ---------------------------------------------------------------------------
__global__ void __launch_bounds__(512)
vq_prep(const float* __restrict__ w, float* __restrict__ w2,
        unsigned int* __restrict__ hist, float* __restrict__ sse,
        float4* __restrict__ wpack) {
  const int t    = blockIdx.x * 512 + threadIdx.x;   // 0..8191
  const int lane = t & 31;
  const int rest = t >> 5;                           // ct*8 + sp, 0..255
  const int sp   = rest & 7;
  const int ct   = rest >> 3;
  const int half = lane >> 4, mlow = lane & 15, koff = 2 * half;
  const int kcol = ct * 16 + mlow;
  float4 f;
  f.x = w[(8 * sp + koff    ) * KCB + kcol];
  f.y = w[(8 * sp + koff + 1) * KCB + kcol];
  f.z = w[(8 * sp + 4 + koff    ) * KCB + kcol];
  f.w = w[(8 * sp + 4 + koff + 1) * KCB + kcol];
  wpack[rest * 32 + lane] = f;

  if (blockIdx.x == 0) {
    int k = threadIdx.x;                             // 0..511
    float s = 0.f;
#pragma unroll 4
    for (int d = 0; d < DIM; ++d) { float v = w[d * KCB + k]; s += v * v; }
    w2[k] = s;
    hist[k] = 0u;
    if (k == 0) *sse = 0.f;
  }
}

// ---------------------------------------------------------------------------
// TDM helper: DMA one contiguous 16 KB chunk (2048 x 8B elements) into LDS.
// D# group0: count=1 | lds_addr | global_addr(57b) | type=2.
// D# group1: data_size=3 (8B); tensor_dim0=tile_dim0=stride=2048; dim1=1.
// ---------------------------------------------------------------------------
#if HAVE_TDM
__device__ __forceinline__ void tdm_load_16k(unsigned lds_off, const void* gptr) {
  unsigned long long ga = (unsigned long long)(size_t)gptr;
  u32x4 g0 = { 1u,                                      // count=1, user mode
               lds_off,                                 // LDS byte address
               (unsigned)(ga & 0xFFFFFFFFu),            // global_addr[31:0]
               (unsigned)((ga >> 32) & 0x1FFFFFFu) | (2u << 30) }; // [56:32]|type=2
  i32x8 g1 = { 0x00030000,        // data_size=3 (8 bytes)
               0x08000000,        // tensor_dim0[15:0]=2048 in bits[63:48]
               0x00010000,        // tensor_dim1=1 in bits[95:80]
               0x08000000,        // tile_dim0=2048 in bits[127:112]
               0x00000001,        // tile_dim1=1
               2048,              // tensor_dim0_stride[31:0]
               0, 0 };
  i32x4 gz = { 0, 0, 0, 0 };
#if __clang_major__ >= 23
  i32x8 gz8 = { 0, 0, 0, 0, 0, 0, 0, 0 };
  __builtin_amdgcn_tensor_load_to_lds(g0, g1, gz, gz, gz8, 0);
#else
  __builtin_amdgcn_tensor_load_to_lds(g0, g1, gz, gz, 0);
#endif
}
#endif

// ---------------------------------------------------------------------------
// Kernel 1: WMMA f32 distances + per-row argmin over K=512.
// One wave owns 16 rows; D=64 -> 16 chained v_wmma_f32_16x16x4_f32 per
// 16-code tile. Codebook (packed) staged into LDS by the TDM, double-buffered
// 16 KB chunks; B fragments read with ds_load_b128.
// ---------------------------------------------------------------------------
__global__ void __launch_bounds__(32 * WAVES_PER_BLOCK)
vq_argmin(const float* __restrict__ x, const float4* __restrict__ wpack,
          const float* __restrict__ w2, int* __restrict__ idx_out,
          float* __restrict__ idx_f_out, unsigned int* __restrict__ hist) {
  const int lane    = threadIdx.x & 31;
  const int wave    = threadIdx.x >> 5;
  const int rowBase = blockIdx.x * (ROWS_PER_WAVE * WAVES_PER_BLOCK)
                    + wave * ROWS_PER_WAVE;
  const int half = lane >> 4;   // A: lanes 0-15 hold K={0,1}, lanes 16-31 K={2,3}
  const int mlow = lane & 15;   // M (row) for A ; N (col) for B/C/D
  const int koff = half * 2;

  // Preload A fragments: 16 k-slices of width 4 for row rowBase+mlow.
  v2f a[16];
  const float* xrow = x + (size_t)(rowBase + mlow) * DIM;
#pragma unroll
  for (int s = 0; s < 16; ++s)
    a[s] = *(const v2f*)(xrow + 4 * s + koff);

  float best[8];
  int   bidx[8];
#pragma unroll
  for (int r = 0; r < 8; ++r) { best[r] = 3.4e38f; bidx[r] = 0; }

#if HAVE_TDM
  __shared__ alignas(16) float4 sb[2][CHUNK_F4];   // 2 x 16 KB double buffer

  if (wave == 0) tdm_load_16k((unsigned)(size_t)&sb[0][0], wpack);

  for (int c = 0; c < NCHUNKS; ++c) {
    if (wave == 0) {
      if (c + 1 < NCHUNKS) {
        tdm_load_16k((unsigned)(size_t)&sb[(c + 1) & 1][0],
                     wpack + (size_t)(c + 1) * CHUNK_F4);
        __builtin_amdgcn_s_wait_tensorcnt(1);   // chunk c done (in-order)
      } else {
        __builtin_amdgcn_s_wait_tensorcnt(0);   // last chunk done
      }
    }
    __syncthreads();                       // chunk c resident in LDS

    const float4* bt = &sb[c & 1][0];
#pragma unroll
    for (int tt = 0; tt < CHUNK_TILES; ++tt) {
      const int ct   = c * CHUNK_TILES + tt;
      const int kcol = ct * 16 + mlow;
      v2f b[16];
#pragma unroll
      for (int sp = 0; sp < 8; ++sp) {     // 8 x ds_load_b128 per tile
        float4 f = bt[(tt * 8 + sp) * 32 + lane];
        b[2 * sp    ] = v2f{ f.x, f.y };
        b[2 * sp + 1] = v2f{ f.z, f.w };
      }
      v8f cacc = {};
#pragma unroll
      for (int s = 0; s < 16; ++s)
        cacc = __builtin_amdgcn_wmma_f32_16x16x4_f32(
                   false, a[s], false, b[s], (short)0, cacc, false, false);
      const float wk = w2[kcol];
#pragma unroll
      for (int r = 0; r < 8; ++r) {
        float dv = fmaf(-2.0f, cacc[r], wk);   // |x|^2 constant/row -> dropped
        if (dv < best[r]) { best[r] = dv; bidx[r] = kcol; }
      }
    }
    __syncthreads();                       // done reading buf before reuse
  }
#else
  for (int ct = 0; ct < NTILES; ++ct) {
    const int kcol = ct * 16 + mlow;
    v2f b[16];
#pragma unroll
    for (int sp = 0; sp < 8; ++sp) {       // 8 x global_load_b128 per tile
      float4 f = wpack[(ct * 8 + sp) * 32 + lane];
      b[2 * sp    ] = v2f{ f.x, f.y };
      b[2 * sp + 1] = v2f{ f.z, f.w };
    }
    v8f cacc = {};
#pragma unroll
    for (int s = 0; s < 16; ++s)
      cacc = __builtin_amdgcn_wmma_f32_16x16x4_f32(
                 false, a[s], false, b[s], (short)0, cacc, false, false);
    const float wk = w2[kcol];
#pragma unroll
    for (int r = 0; r < 8; ++r) {
      float dv = fmaf(-2.0f, cacc[r], wk);
      if (dv < best[r]) { best[r] = dv; bidx[r] = kcol; }
    }
  }
#endif

  // Min-reduce across the 16 lanes of each half (xor masks 1..8 stay in-half).
#pragma unroll
  for (int m = 8; m >= 1; m >>= 1) {
#pragma unroll
    for (int r = 0; r < 8; ++r) {
      float ov = __shfl_xor(best[r], m, 32);
      int   oi = __shfl_xor(bidx[r], m, 32);
      if (ov < best[r] || (ov == best[r] && oi < bidx[r])) {
        best[r] = ov; bidx[r] = oi;
      }
    }
  }

  if (mlow == 0) {                // lane 0 -> rows 0..7 ; lane 16 -> rows 8..15
#pragma unroll
    for (int r = 0; r < 8; ++r) {
      int row = rowBase + half * 8 + r;
      idx_out[row]   = bidx[r];
      idx_f_out[row] = (float)bidx[r];
      atomicAdd(&hist[bidx[r]], 1u);
    }
  }
}

// ---------------------------------------------------------------------------
// Kernel 2: gather codebook rows into quantized output + fused SSE reduction.
// ---------------------------------------------------------------------------
__global__ void __launch_bounds__(256)
vq_gather(const float* __restrict__ x, const float* __restrict__ w,
          const int* __restrict__ idx, float* __restrict__ out_q,
          float* __restrict__ sse) {
  const int i = blockIdx.x * 256 + threadIdx.x;   // element id, < N*D = 4M
  const int n = i >> 6;
  const int d = i & 63;
  const int k = idx[n];
  float q = w[d * KCB + k];
  out_q[i] = q;
  float diff = q - x[i];
  float v = diff * diff;
#pragma unroll
  for (int m = 16; m >= 1; m >>= 1) v += __shfl_xor(v, m, 32);
  if ((threadIdx.x & 31) == 0) atomicAdd(sse, v);
}

// ---------------------------------------------------------------------------
// Kernel 3: loss = 1.25 * SSE / (N*D); perplexity = exp(-sum p log(p+1e-10)).
// ---------------------------------------------------------------------------
__global__ void __launch_bounds__(512)
vq_finalize(const unsigned int* __restrict__ hist, const float* __restrict__ sse,
            float* __restrict__ out_scalars) {
  __shared__ float red[16];
  int k = threadIdx.x;                       // 512 threads = 16 waves
  float p = (float)hist[k] * (1.0f / (float)NROWS);
  float term = p * logf(p + 1e-10f);
#pragma unroll
  for (int m = 16; m >= 1; m >>= 1) term += __shfl_xor(term, m, 32);
  if ((k & 31) == 0) red[k >> 5] = term;
  __syncthreads();
  if (k < 16) {
    float t = red[k];
#pragma unroll
    for (int m = 8; m >= 1; m >>= 1) t += __shfl_xor(t, m, 32);
    if (k == 0) {
      out_scalars[0] = 1.25f * (*sse) / (float)((size_t)NROWS * DIM); // loss
      out_scalars[1] = expf(-t);                                      // perplexity
    }
  }
}

// ---------------------------------------------------------------------------
extern "C" void kernel_launch(void* const* d_in, const int* in_sizes, int n_in,
                              void* d_out, int out_size, void* d_ws, size_t ws_size,
                              hipStream_t stream) {
  (void)in_sizes; (void)n_in; (void)out_size; (void)ws_size;
  const float* x = (const float*)d_in[0];   // [64,32,32,64] fp32
  const float* w = (const float*)d_in[1];   // [64,512] fp32

  float* out         = (float*)d_out;
  float* out_q       = out;                              // N*D quantized_st
  float* out_scalars = out + (size_t)NROWS * DIM;        // loss, perplexity
  float* out_idx     = out + (size_t)NROWS * DIM + 2;    // N indices (as float)

  // Packed codebook scratch lives in the head of d_out (128 KB); it is fully
  // consumed by vq_argmin and then overwritten by vq_gather's output.
  float4* wpack = (float4*)d_out;

  char*         ws   = (char*)d_ws;
  float*        w2   = (float*)ws;                       // 512 f32
  unsigned int* hist = (unsigned int*)(ws + 2048);       // 512 u32
  float*        sse  = (float*)(ws + 4096);              // 1 f32
  int*          idx  = (int*)(ws + 4160);                // N i32

  vq_prep<<<16, 512, 0, stream>>>(w, w2, hist, sse, wpack);
  vq_argmin<<<NROWS / (ROWS_PER_WAVE * WAVES_PER_BLOCK),
              32 * WAVES_PER_BLOCK, 0, stream>>>(x, wpack, w2, idx, out_idx, hist);
  vq_gather<<<((size_t)NROWS * DIM) / 256, 256, 0, stream>>>(x, w, idx, out_q, sse);
  vq_finalize<<<1, 512, 0, stream>>>(hist, sse, out_scalars);
}